// HeteroGNN_concat_5403068858514
// MI455X (gfx1250) — compile-verified
//
#include <hip/hip_runtime.h>

typedef __attribute__((ext_vector_type(2))) float v2f;
typedef __attribute__((ext_vector_type(8))) float v8f;

#define H 64
#define F85 85
#define KPAD 88

// ---------------------------------------------------------------------------
// Utility kernels
// ---------------------------------------------------------------------------
__global__ void zero_kernel(float* __restrict__ p, long n) {
    long i = (long)blockIdx.x * blockDim.x + threadIdx.x;
    if (i < n) p[i] = 0.0f;
}

// Build padded 88x64 weight blocks for conv1 GEMMs + summed transaction bias.
__global__ void prep_w_kernel(const float* __restrict__ w1_src_t,  // [2][1][64]
                              const float* __restrict__ w1_dst_t,  // [2][85][64]
                              const float* __restrict__ b1_t,      // [2][64]
                              const float* __restrict__ w1_src_r,  // [2][85][64]
                              const float* __restrict__ w1_dst_r,  // [2][1][64]
                              float* __restrict__ W1t, float* __restrict__ W1c,
                              float* __restrict__ W1m, float* __restrict__ b1sum) {
    int i = blockIdx.x * blockDim.x + threadIdx.x;
    if (i >= KPAD * H) return;
    int r = i >> 6, c = i & 63;
    float vt = 0.f, vc = 0.f, vm = 0.f;
    if (r < F85) {
        vt = w1_dst_t[r * H + c] + w1_dst_t[F85 * H + r * H + c];
        vc = w1_src_r[r * H + c];
        vm = w1_src_r[F85 * H + r * H + c];
    } else if (r == 85) {
        vt = w1_src_t[c];            // card scalar -> trans
        vc = w1_dst_r[c];            // root transform of x_card
        vm = w1_dst_r[H + c];        // root transform of x_merch
    } else if (r == 86) {
        vt = w1_src_t[H + c];        // merch scalar -> trans
    }
    W1t[i] = vt; W1c[i] = vc; W1m[i] = vm;
    if (i < H) b1sum[i] = b1_t[i] + b1_t[H + i];
}

// conv1 forward scalar aggregation: agg[dst] += x[src]
__global__ void scatter_scalar_kernel(const float* __restrict__ xc, const float* __restrict__ xm,
                                      const int* __restrict__ sp, const int* __restrict__ dp,
                                      const int* __restrict__ sr, const int* __restrict__ dr,
                                      float* __restrict__ aggc, float* __restrict__ aggm, int E) {
    int e = blockIdx.x * blockDim.x + threadIdx.x;
    if (e >= E) return;
    atomicAdd(&aggc[dp[e]], xc[sp[e]]);
    atomicAdd(&aggm[dr[e]], xm[sr[e]]);
}

// generic F-wide feature scatter-add: out[oidx[e]][j] += in[iidx[e]][j]
__global__ void scatter_feat_kernel(float* __restrict__ outbuf, const int* __restrict__ oidx,
                                    const float* __restrict__ inbuf, const int* __restrict__ iidx,
                                    int E, int F) {
    long i = (long)blockIdx.x * blockDim.x + threadIdx.x;
    long total = (long)E * F;
    if (i >= total) return;
    int e = (int)(i / F), j = (int)(i % F);
    atomicAdd(&outbuf[(size_t)oidx[e] * F + j], inbuf[(size_t)iidx[e] * F + j]);
}

// ---------------------------------------------------------------------------
// Generic conv1 GEMM: out[M x 64] = relu( [Xmain(85) | e1 | e2 | 0] @ W[88x64] + bias )
// blockDim = 128 (4 waves); each wave owns a 16-wide column tile; 16 rows/block.
// ---------------------------------------------------------------------------
__global__ __launch_bounds__(128) void gemm88_kernel(
    const float* __restrict__ Xmain, const float* __restrict__ extra1,
    const float* __restrict__ extra2, const float* __restrict__ W,
    const float* __restrict__ bias, float* __restrict__ out, int M) {
    __shared__ float xs[16 * KPAD];

    const int row0 = blockIdx.x * 16;
    // cooperative stage of padded 16x88 X tile into LDS
    for (int i = threadIdx.x; i < 16 * KPAD; i += 128) {
        int r = i / KPAD, c = i - r * KPAD;
        int grow = row0 + r;
        float v;
        if (c < F85)      v = Xmain[(size_t)grow * F85 + c];
        else if (c == 85) v = extra1 ? extra1[grow] : 0.0f;
        else if (c == 86) v = extra2 ? extra2[grow] : 0.0f;
        else              v = 0.0f;
        xs[i] = v;
    }
    __syncthreads();

    const int wave = threadIdx.x >> 5;
    const int lane = threadIdx.x & 31;
    const int half = lane >> 4;   // K pair selector for A/B fragments
    const int lm   = lane & 15;   // A: row in tile; B/C: column in tile
    const int c0   = wave * 16;

    const float bval = bias[c0 + lm];
    v8f acc;
#pragma unroll
    for (int i = 0; i < 8; ++i) acc[i] = bval;

    for (int k = 0; k < KPAD; k += 4) {
        const int ka = k + 2 * half;
        v2f a, b;
        a.x = xs[lm * KPAD + ka];
        a.y = xs[lm * KPAD + ka + 1];
        b.x = W[(size_t)(ka + 0) * H + c0 + lm];
        b.y = W[(size_t)(ka + 1) * H + c0 + lm];
        acc = __builtin_amdgcn_wmma_f32_16x16x4_f32(false, a, false, b, (short)0, acc,
                                                    false, false);
    }
    // relu + store: lane holds rows v + 8*half, column c0+lm
#pragma unroll
    for (int v = 0; v < 8; ++v) {
        float r = acc[v];
        r = r > 0.0f ? r : 0.0f;
        out[(size_t)(row0 + v + 8 * half) * H + c0 + lm] = r;
    }
}

// ---------------------------------------------------------------------------
// Fused conv2 + concat + final Linear(128,2).
// Per 16-row transaction tile: each wave accumulates T1 (pays) and T2 (recv)
// 16x16 tiles (4 wmma per k-step), relu -> LDS 16x128 concat -> 128x2 linear.
// ---------------------------------------------------------------------------
__global__ __launch_bounds__(128) void conv2_out_kernel(
    const float* __restrict__ h_t,  const float* __restrict__ aggc,
    const float* __restrict__ aggm, const float* __restrict__ w2l,
    const float* __restrict__ w2r,  const float* __restrict__ b2,
    const float* __restrict__ wlin, const float* __restrict__ blin,
    float* __restrict__ out, int M) {
    __shared__ float hs[16 * H];
    __shared__ float cs[16 * H];
    __shared__ float ms[16 * H];
    __shared__ float ts[16 * 2 * H];

    const int row0 = blockIdx.x * 16;
    for (int i = threadIdx.x; i < 16 * H; i += 128) {
        int r = i >> 6, c = i & 63;
        size_t g = (size_t)(row0 + r) * H + c;
        hs[i] = h_t[g];
        cs[i] = aggc[g];
        ms[i] = aggm[g];
    }
    __syncthreads();

    const int wave = threadIdx.x >> 5;
    const int lane = threadIdx.x & 31;
    const int half = lane >> 4;
    const int lm   = lane & 15;
    const int c0   = wave * 16;

    v8f acc1, acc2;
    const float bv1 = b2[0 * H + c0 + lm];
    const float bv2 = b2[1 * H + c0 + lm];
#pragma unroll
    for (int i = 0; i < 8; ++i) { acc1[i] = bv1; acc2[i] = bv2; }

    const float* W2l0 = w2l;                 // pays
    const float* W2l1 = w2l + 1 * H * H;     // receives
    const float* W2r0 = w2r;
    const float* W2r1 = w2r + 1 * H * H;

    for (int k = 0; k < H; k += 4) {
        const int ka = k + 2 * half;
        v2f ah, ac, am, b;
        ah.x = hs[lm * H + ka]; ah.y = hs[lm * H + ka + 1];
        ac.x = cs[lm * H + ka]; ac.y = cs[lm * H + ka + 1];
        am.x = ms[lm * H + ka]; am.y = ms[lm * H + ka + 1];

        b.x = W2l0[(ka + 0) * H + c0 + lm]; b.y = W2l0[(ka + 1) * H + c0 + lm];
        acc1 = __builtin_amdgcn_wmma_f32_16x16x4_f32(false, ac, false, b, (short)0, acc1,
                                                     false, false);
        b.x = W2r0[(ka + 0) * H + c0 + lm]; b.y = W2r0[(ka + 1) * H + c0 + lm];
        acc1 = __builtin_amdgcn_wmma_f32_16x16x4_f32(false, ah, false, b, (short)0, acc1,
                                                     false, false);
        b.x = W2l1[(ka + 0) * H + c0 + lm]; b.y = W2l1[(ka + 1) * H + c0 + lm];
        acc2 = __builtin_amdgcn_wmma_f32_16x16x4_f32(false, am, false, b, (short)0, acc2,
                                                     false, false);
        b.x = W2r1[(ka + 0) * H + c0 + lm]; b.y = W2r1[(ka + 1) * H + c0 + lm];
        acc2 = __builtin_amdgcn_wmma_f32_16x16x4_f32(false, ah, false, b, (short)0, acc2,
                                                     false, false);
    }

#pragma unroll
    for (int v = 0; v < 8; ++v) {
        float r1 = acc1[v]; r1 = r1 > 0.0f ? r1 : 0.0f;
        float r2 = acc2[v]; r2 = r2 > 0.0f ? r2 : 0.0f;
        ts[(v + 8 * half) * 128 + c0 + lm]      = r1;   // t1 -> cols 0..63
        ts[(v + 8 * half) * 128 + 64 + c0 + lm] = r2;   // t2 -> cols 64..127
    }
    __syncthreads();

    if (threadIdx.x < 32) {
        const int r = threadIdx.x >> 1, c = threadIdx.x & 1;
        float s = blin[c];
#pragma unroll 8
        for (int j = 0; j < 128; ++j) s += ts[r * 128 + j] * wlin[j * 2 + c];
        out[(size_t)(row0 + r) * 2 + c] = s;
    }
}

// ---------------------------------------------------------------------------
extern "C" void kernel_launch(void* const* d_in, const int* in_sizes, int n_in,
                              void* d_out, int out_size, void* d_ws, size_t ws_size,
                              hipStream_t stream) {
    const float* x_card   = (const float*)d_in[0];
    const float* x_merch  = (const float*)d_in[1];
    const float* x_trans  = (const float*)d_in[2];
    const float* w1_src_t = (const float*)d_in[3];
    const float* w1_dst_t = (const float*)d_in[4];
    const float* b1_t     = (const float*)d_in[5];
    const float* w1_src_r = (const float*)d_in[6];
    const float* w1_dst_r = (const float*)d_in[7];
    const float* b1_r     = (const float*)d_in[8];
    const float* w2_l     = (const float*)d_in[9];
    const float* w2_r     = (const float*)d_in[10];
    const float* b2       = (const float*)d_in[11];
    const float* w_lin    = (const float*)d_in[12];
    const float* b_lin    = (const float*)d_in[13];
    const int* src_pays   = (const int*)d_in[14];
    const int* dst_pays   = (const int*)d_in[15];
    const int* src_recv   = (const int*)d_in[16];
    const int* dst_recv   = (const int*)d_in[17];
    float* out = (float*)d_out;

    const int N_c = in_sizes[0];
    const int N_m = in_sizes[1];
    const int N_t = in_sizes[2] / F85;
    const int E   = in_sizes[14];

    // workspace layout (floats)
    float* ws = (float*)d_ws;
    size_t off = 0;
    float* aggc1  = ws + off; off += (size_t)N_t;            // scalar card->trans
    float* aggm1  = ws + off; off += (size_t)N_t;            // scalar merch->trans
    float* aggt2c = ws + off; off += (size_t)N_c * F85;      // trans feats -> card
    float* aggt2m = ws + off; off += (size_t)N_m * F85;      // trans feats -> merch
    float* agg2c  = ws + off; off += (size_t)N_t * H;        // h_c -> trans
    float* agg2m  = ws + off; off += (size_t)N_t * H;        // h_m -> trans
    const long zeroCount = (long)off;
    float* h_t    = ws + off; off += (size_t)N_t * H;
    float* h_c    = ws + off; off += (size_t)N_c * H;
    float* h_m    = ws + off; off += (size_t)N_m * H;
    float* W1t    = ws + off; off += (size_t)KPAD * H;
    float* W1c    = ws + off; off += (size_t)KPAD * H;
    float* W1m    = ws + off; off += (size_t)KPAD * H;
    float* b1sum  = ws + off; off += (size_t)H;

    // 1) zero accumulation regions
    {
        long blocks = (zeroCount + 255) / 256;
        zero_kernel<<<(unsigned)blocks, 256, 0, stream>>>(ws, zeroCount);
    }
    // 2) padded weights
    prep_w_kernel<<<(KPAD * H + 255) / 256, 256, 0, stream>>>(
        w1_src_t, w1_dst_t, b1_t, w1_src_r, w1_dst_r, W1t, W1c, W1m, b1sum);
    // 3) conv1 forward scalar aggregation
    scatter_scalar_kernel<<<(E + 255) / 256, 256, 0, stream>>>(
        x_card, x_merch, src_pays, dst_pays, src_recv, dst_recv, aggc1, aggm1, E);
    // 4) conv1 reverse: x_trans -> card / merch  (85-wide)
    {
        long total = (long)E * F85;
        scatter_feat_kernel<<<(unsigned)((total + 255) / 256), 256, 0, stream>>>(
            aggt2c, src_pays, x_trans, dst_pays, E, F85);
        scatter_feat_kernel<<<(unsigned)((total + 255) / 256), 256, 0, stream>>>(
            aggt2m, src_recv, x_trans, dst_recv, E, F85);
    }
    // 5) conv1 GEMMs (WMMA)
    gemm88_kernel<<<N_t / 16, 128, 0, stream>>>(x_trans, aggc1, aggm1, W1t, b1sum, h_t, N_t);
    gemm88_kernel<<<N_c / 16, 128, 0, stream>>>(aggt2c, x_card, nullptr, W1c, b1_r, h_c, N_c);
    gemm88_kernel<<<N_m / 16, 128, 0, stream>>>(aggt2m, x_merch, nullptr, W1m, b1_r + H, h_m, N_m);
    // 6) conv2 aggregation: h_c/h_m -> trans (64-wide)
    {
        long total = (long)E * H;
        scatter_feat_kernel<<<(unsigned)((total + 255) / 256), 256, 0, stream>>>(
            agg2c, dst_pays, h_c, src_pays, E, H);
        scatter_feat_kernel<<<(unsigned)((total + 255) / 256), 256, 0, stream>>>(
            agg2m, dst_recv, h_m, src_recv, E, H);
    }
    // 7) fused conv2 + concat + final linear (WMMA)
    conv2_out_kernel<<<N_t / 16, 128, 0, stream>>>(
        h_t, agg2c, agg2m, w2_l, w2_r, b2, w_lin, b_lin, out, N_t);
}